// Hydra_10075993276635
// MI455X (gfx1250) — compile-verified
//
#include <hip/hip_runtime.h>

// Mamba selective scan, chunked-SSD formulation for CDNA5 (gfx1250).
// ~48MB traffic -> ~2us roofline @ 23.3TB/s. Per 16-step chunk the scan is a
// 16x16x16 f32 matmul (4x v_wmma_f32_16x16x4_f32) plus lane-group reductions.
// (1) all 16-lane shuffles via v_permlane16_b32 (VALU) instead of ds_bpermute,
// (2) software-pipelined async staging via GLOBAL_LOAD_ASYNC_TO_LDS_B32
//     (builtin signature: (AS1 int* src, AS3 int* dst, imm offset, imm cpol))
//     + s_wait_asynccnt before the publishing barrier.

#define BATCH_   2
#define LSEQ     2048
#define EDIM     1024
#define NST      16            // state dim N
#define TCH      16            // chunk length (time steps)
#define NCH      (LSEQ / TCH)  // 128 chunks
#define EDT      8             // ed channels per block (one wave each)
#define NTHREADS (EDT * 32)    // 256 threads = 8 waves (wave32)

typedef __attribute__((ext_vector_type(2))) float v2f;
typedef __attribute__((ext_vector_type(8))) float v8f;

// ---------------- 16-lane-row shuffles via v_permlane16_b32 ----------------
// v_permlane16_b32: dst lane i reads src lane sel[i] within each 16-lane row;
// selectors are 16x4-bit packed into two 32-bit scalars (runtime SGPRs OK).
#if __has_builtin(__builtin_amdgcn_permlane16)
#define HAVE_PERMLANE16 1
__device__ __forceinline__ float permf16(float v, unsigned s0, unsigned s1) {
  return __int_as_float(__builtin_amdgcn_permlane16(
      __float_as_int(v), __float_as_int(v), (int)s0, (int)s1, false, false));
}
// xor-butterfly selectors, index i -> offset (1<<i)
__device__ constexpr unsigned XS0[4] = {0x67452301u, 0x54761032u, 0x32107654u, 0xFEDCBA98u};
__device__ constexpr unsigned XS1[4] = {0xEFCDAB89u, 0xDCFE98BAu, 0xBA98FEDCu, 0x76543210u};
// shfl_up selectors, sel[i] = (i - (1<<i_log)) & 15 (wrap; caller masks)
__device__ constexpr unsigned US0[4] = {0x6543210Fu, 0x543210FEu, 0x3210FEDCu, 0xFEDCBA98u};
__device__ constexpr unsigned US1[4] = {0xEDCBA987u, 0xDCBA9876u, 0xBA987654u, 0x76543210u};
#define GRP_XOR(v, i) permf16((v), XS0[i], XS1[i])
#define GRP_UP(v, i)  permf16((v), US0[i], US1[i])
#define GRP_B15(v)    permf16((v), 0xFFFFFFFFu, 0xFFFFFFFFu)
#else
#define HAVE_PERMLANE16 0
#define GRP_XOR(v, i) __shfl_xor((v), 1 << (i), 16)
#define GRP_UP(v, i)  __shfl_up((v), 1u << (i), 16)
#define GRP_B15(v)    __shfl((v), 15, 16)
#endif

// ---------------- async global->LDS staging (fire-and-forget) ----------------
#if __has_builtin(__builtin_amdgcn_global_load_async_to_lds_b32)
#define USE_ASYNC 1
typedef __attribute__((address_space(1))) int GInt;   // prints as '__device__ int *'
typedef __attribute__((address_space(3))) int LInt;   // '__shared__ int *'
// flat shared-aperture pointers: low 32 bits are the LDS byte offset
#define AS1P(p) ((GInt*)(unsigned long long)(const void*)(p))
#define AS3P(p) ((LInt*)(unsigned)(unsigned long long)(const void*)(p))
#define ASYNC_B32(gsrc, ldst) \
  __builtin_amdgcn_global_load_async_to_lds_b32(AS1P(gsrc), AS3P(ldst), 0, 0)
#if __has_builtin(__builtin_amdgcn_s_wait_asynccnt)
#define WAIT_ASYNC0() __builtin_amdgcn_s_wait_asynccnt(0)
#else
#define WAIT_ASYNC0() asm volatile("s_wait_asynccnt 0x0" ::: "memory")
#endif
#else
#define USE_ASYNC 0
#define WAIT_ASYNC0()
#endif

__global__ __launch_bounds__(NTHREADS)
void mamba_scan_wmma(const float* __restrict__ Xg,    // (b, l, ed)
                     const float* __restrict__ DLg,   // delta (b, l, ed)
                     const float* __restrict__ Ag,    // (ed, n)
                     const float* __restrict__ Bg,    // (b, l, n)
                     const float* __restrict__ Cg,    // (b, l, n)
                     const float* __restrict__ Dg,    // (ed,)
                     float* __restrict__ Yg) {        // (b, l, ed)
  // Padded LDS tiles (pad keeps strided reads bank-conflict free on 64 banks).
  __shared__ float sB [2][TCH][NST + 1];
  __shared__ float sC [2][TCH][NST + 1];
  __shared__ float sX [2][TCH][EDT + 1];
  __shared__ float sDl[2][TCH][EDT + 1];
  __shared__ float sY    [TCH][EDT + 1];

  const int tid  = threadIdx.x;
  const int wave = tid >> 5;        // 0..7  -> local ed channel
  const int lane = tid & 31;
  const int half = lane >> 4;       // 0: n in {4k,4k+1}, 1: n in {4k+2,4k+3}
  const int tq   = lane & 15;       // this lane's time index t (A op) / s (B op)

  const int bb  = blockIdx.x / (EDIM / EDT);
  const int ed0 = (blockIdx.x % (EDIM / EDT)) * EDT;
  const int ed  = ed0 + wave;

  // Per-lane A[ed, n] for the 8 n-values this lane owns in BOTH wmma operands:
  // slice k, reg j -> n = 4k + 2*half + j (16x4 f32 A layout == 4x16 f32 B layout).
  float a_reg[4][2];
#pragma unroll
  for (int k = 0; k < 4; ++k)
#pragma unroll
    for (int j = 0; j < 2; ++j)
      a_reg[k][j] = Ag[ed * NST + 4 * k + 2 * half + j];

  const float Dv = Dg[ed];

  // Hidden state; live only in lanes 0/16 (the N==0 column owners of the B op).
  float h_reg[4][2] = {};

  // Cooperative-load index precompute.
  const int bt = tid >> 4, bn = tid & 15;   // B/C tile: 256 contiguous floats
  const int i128 = tid & 127;
  const int xt = i128 >> 3, xe = i128 & 7;  // x/delta tile: 16 t x 8 ed
  const int xg_base = ((bb * LSEQ + xt) * EDIM) + ed0 + xe;
  const int bc_base = (bb * LSEQ) * NST + tid;

  // Stage chunk c's tiles into LDS buffer (c&1); async when available.
  auto stage = [&](int c) {
    const int sbuf = c & 1;
    const int bci  = bc_base + c * TCH * NST;   // B/C rows fully contiguous
    const int gi   = xg_base + c * TCH * EDIM;  // 32B rows, 4KB stride
#if USE_ASYNC
    ASYNC_B32(Bg + bci, &sB[sbuf][bt][bn]);
    ASYNC_B32(Cg + bci, &sC[sbuf][bt][bn]);
    if (tid < 128) ASYNC_B32(Xg + gi,  &sX [sbuf][xt][xe]);
    else           ASYNC_B32(DLg + gi, &sDl[sbuf][xt][xe]);
#else
    sB[sbuf][bt][bn] = Bg[bci];
    sC[sbuf][bt][bn] = Cg[bci];
    if (tid < 128) sX [sbuf][xt][xe] = Xg[gi];
    else           sDl[sbuf][xt][xe] = DLg[gi];
#endif
    if (c + 1 < NCH) {  // distance-2 prefetch (global_prefetch_b8)
      __builtin_prefetch(Bg + bci + TCH * NST, 0, 1);
      __builtin_prefetch(Cg + bci + TCH * NST, 0, 1);
      __builtin_prefetch((tid < 128 ? Xg : DLg) + gi + TCH * EDIM, 0, 1);
    }
  };

  stage(0);
  for (int c = 0; c < NCH; ++c) {
    const int l0  = c * TCH;
    const int buf = c & 1;

    WAIT_ASYNC0();        // my async tile-stores to LDS have landed
    __syncthreads();      // everyone's have landed -> tiles for chunk c ready
    if (c + 1 < NCH) stage(c + 1);  // overlap next chunk's copy with compute

    // ---- per-lane chunk math ----
    const float d0 = sDl[buf][tq][wave];
    const float xv = sX [buf][tq][wave];

    // Inclusive cumsum of delta over the 16-lane row: S = S_{tq}.
    float S = d0;
#pragma unroll
    for (int i = 0; i < 4; ++i) {
      float u = GRP_UP(S, i);
      if (tq >= (1 << i)) S += u;
    }
    const float S15 = GRP_B15(S);
    const float dx  = d0 * xv;

    // Cb (A op) and Bb^T (B op) slices; 4 chained WMMAs:
    //   M[t,s] = sum_n Cb[t,n] * Bb[s,n]    (16x16x16 f32)
    v8f acc = {};
    v2f bsv[4];
#pragma unroll
    for (int k = 0; k < 4; ++k) {
      v2f af, bfv;
#pragma unroll
      for (int j = 0; j < 2; ++j) {
        const int n    = 4 * k + 2 * half + j;
        const float aS = a_reg[k][j] * S;
        float cb  = sC[buf][tq][n] * __expf(aS);        // C[t,n]*e^{A S_t}
        float bbv = sB[buf][tq][n] * dx * __expf(-aS);  // B[s,n]*d_s*x_s*e^{-A S_s}
        if (tq == 0) bbv += h_reg[k][j];                // fold h_prev into s=0 row
        af[j]  = cb;
        bfv[j] = bbv;
      }
      bsv[k] = bfv;
      acc = __builtin_amdgcn_wmma_f32_16x16x4_f32(
          /*neg_a=*/false, af, /*neg_b=*/false, bfv,
          /*c_mod=*/(short)0, acc, /*reuse_a=*/false, /*reuse_b=*/false);
    }

    // ---- state update: h_new[n] = e^{A_n S_15} * colsum_s(Bb) ----
#pragma unroll
    for (int k = 0; k < 4; ++k)
#pragma unroll
      for (int j = 0; j < 2; ++j) {
        float cs = bsv[k][j];
#pragma unroll
        for (int i = 3; i >= 0; --i) cs += GRP_XOR(cs, i);
        h_reg[k][j] = __expf(a_reg[k][j] * S15) * cs;
      }

    // ---- y_t = sum_{s<=t} M[t,s] + D*x_t  (masked row-sum of D tile) ----
    // D layout: VGPR r -> row t = r + 8*half, column s = tq.
    const float xw  = sX[buf][(tq & 7) + 8 * half][wave];
    const float dxw = Dv * xw;
#pragma unroll
    for (int r = 0; r < 8; ++r) {
      float m = (tq <= r + 8 * half) ? acc[r] : 0.0f;
#pragma unroll
      for (int i = 3; i >= 0; --i) m += GRP_XOR(m, i);
      if (tq == r) sY[r + 8 * half][wave] = m + dxw;
    }
    __syncthreads();

    // ---- cooperative coalesced store of the 16 x 8 y tile ----
    if (tid < 128)
      Yg[((bb * LSEQ + l0 + xt) * EDIM) + ed0 + xe] = sY[xt][xe];
  }
}

extern "C" void kernel_launch(void* const* d_in, const int* in_sizes, int n_in,
                              void* d_out, int out_size, void* d_ws, size_t ws_size,
                              hipStream_t stream) {
  const float* x     = (const float*)d_in[0];
  const float* delta = (const float*)d_in[1];
  const float* A     = (const float*)d_in[2];
  const float* B     = (const float*)d_in[3];
  const float* C     = (const float*)d_in[4];
  const float* D     = (const float*)d_in[5];
  float* y = (float*)d_out;

  dim3 grid(BATCH_ * (EDIM / EDT));   // 256 blocks
  dim3 block(NTHREADS);               // 8 waves
  hipLaunchKernelGGL(mamba_scan_wmma, grid, block, 0, stream,
                     x, delta, A, B, C, D, y);
  (void)in_sizes; (void)n_in; (void)out_size; (void)d_ws; (void)ws_size;
}